// FeaturePropagation_5600637354112
// MI455X (gfx1250) — compile-verified
//
#include <hip/hip_runtime.h>
#include <hip/hip_bf16.h>
#include <stdint.h>

#define NN    50000
#define NG    512
#define NE    800000
#define NITER 5
#define RPB   16            // destination rows per block
#define CHUNK 16            // edges staged per pipeline stage
#define SSTRIDE 528         // stage row stride in floats (512 + 16: half-wave bank decoupling)

// ---- workspace layout (bytes) ------------------------------------------
#define OFF_ROWPTR 0u
#define OFF_CURSOR 204800u
#define OFF_SCOL   409600u
#define OFF_SWT    3609600u
#define OFF_IMP    6809600u
#define OFF_XBUF   32409600u

typedef float v2f __attribute__((ext_vector_type(2)));
typedef float v8f __attribute__((ext_vector_type(8)));

// ---- CDNA5 async global->LDS helpers (inline asm) -----------------------
__device__ __forceinline__ void async_ld_b128(unsigned ldsaddr, unsigned long long gaddr) {
    asm volatile("global_load_async_to_lds_b128 %0, %1, off"
                 :: "v"(ldsaddr), "v"(gaddr) : "memory");
}
#define WAIT_ASYNC(N) asm volatile("s_wait_asynccnt " #N ::: "memory")

// deterministic per-(iteration, element) hash standing in for threefry
__device__ __forceinline__ unsigned rnd_hash(unsigned it, unsigned rg) {
    unsigned h = rg * 0x9E3779B1u + it * 0x85EBCA77u + 0x27D4EB2Fu;
    h ^= h >> 16; h *= 0x7FEB352Du;
    h ^= h >> 15; h *= 0x846CA68Bu;
    h ^= h >> 16;
    if (h == 0xFFFFFFFFu) h = 0xFFFFFFFEu;
    return h;
}

// ---- one-time CSR construction -----------------------------------------
__global__ void k_zero(int* __restrict__ p, size_t n) {
    size_t i = (size_t)blockIdx.x * blockDim.x + threadIdx.x;
    size_t stride = (size_t)gridDim.x * blockDim.x;
    for (; i < n; i += stride) p[i] = 0;
}

__global__ void k_hist(const int* __restrict__ rows, int* __restrict__ cnt, int nE) {
    int e = blockIdx.x * blockDim.x + threadIdx.x;
    if (e < nE) atomicAdd(&cnt[rows[e]], 1);
}

__global__ void k_scan(int* __restrict__ cnt, int* __restrict__ rowPtr, int n) {
    __shared__ int buf[1024];
    __shared__ int base;
    const int t = threadIdx.x;
    if (t == 0) base = 0;
    __syncthreads();
    for (int start = 0; start < n; start += 1024) {
        int i = start + t;
        int v = (i < n) ? cnt[i] : 0;
        buf[t] = v;
        __syncthreads();
        for (int off = 1; off < 1024; off <<= 1) {
            int a = (t >= off) ? buf[t - off] : 0;
            __syncthreads();
            buf[t] += a;
            __syncthreads();
        }
        int excl = buf[t] - v;
        if (i < n) { rowPtr[i] = base + excl; cnt[i] = base + excl; }
        __syncthreads();
        if (t == 0) base += buf[1023];
        __syncthreads();
    }
    if (t == 0) rowPtr[n] = base;
}

__global__ void k_scatter(const int* __restrict__ rows, const int* __restrict__ cols,
                          const float* __restrict__ wts, int* __restrict__ cursor,
                          int* __restrict__ scol, float* __restrict__ swt, int nE) {
    int e = blockIdx.x * blockDim.x + threadIdx.x;
    if (e < nE) {
        int p = atomicAdd(&cursor[rows[e]], 1);
        scol[p] = cols[e];
        swt[p]  = wts[e];
    }
}

// canonical per-row edge order -> deterministic fp accumulation
__global__ void k_sortrows(const int* __restrict__ rowPtr, int* __restrict__ scol,
                           float* __restrict__ swt, int n) {
    int r = blockIdx.x * blockDim.x + threadIdx.x;
    if (r >= n) return;
    int b = rowPtr[r], e = rowPtr[r + 1];
    for (int i = b + 1; i < e; ++i) {
        int c = scol[i]; float w = swt[i];
        unsigned long long key = ((unsigned long long)(unsigned)c << 32) | __float_as_uint(w);
        int j = i - 1;
        while (j >= b) {
            unsigned long long kj =
                ((unsigned long long)(unsigned)scol[j] << 32) | __float_as_uint(swt[j]);
            if (kj <= key) break;
            scol[j + 1] = scol[j]; swt[j + 1] = swt[j]; --j;
        }
        scol[j + 1] = c; swt[j + 1] = w;
    }
}

// ---- WMMA propagate: 16 dest rows per block ----------------------------
// xp(16 rows x 512 genes) = Sum_k  A(16x4 one-hot weights) x B(4x16 gathered genes)
__global__ __launch_bounds__(256) void k_prop(
    const float* __restrict__ xsrc, float* __restrict__ xdst,
    const float* __restrict__ xorig, unsigned char* __restrict__ imp,
    const int* __restrict__ rowPtr, const int* __restrict__ scol,
    const float* __restrict__ swt, const int* __restrict__ caps, int it)
{
    __shared__ float    stageMem[2 * CHUNK * SSTRIDE]; // 67.5KB; reused as xp after loop
    __shared__ unsigned sc[NG];
    __shared__ unsigned hist[256];
    __shared__ int      cumex[256];
    __shared__ int      rp[RPB + 1];
    __shared__ float2   lw2[CHUNK];     // per staged edge: {weight (0 pad), asfloat(local row)}
    __shared__ int      sT, sTotal;

    const int t      = threadIdx.x;
    const int lane   = t & 31;
    const int wv     = t >> 5;              // wave id 0..7
    const int half   = (lane >> 4) & 1;
    const int lm     = lane & 15;
    const int r_base = blockIdx.x * RPB;

    if (t <= RPB) rp[t] = rowPtr[r_base + t];
    __syncthreads();
    const int kbeg = rp[0], kend = rp[RPB];
    const int nChunk = (kend - kbeg + CHUNK - 1) / CHUNK;

    // async stage issue: thread t stages 128B of row (t>>4), segment (t&15)
    const unsigned lds0 = (unsigned)(size_t)&stageMem[0];
    const int srow = t >> 4, sseg = t & 15;

    v8f dsum[4] = {};   // 4 gene tiles per wave: gt = wv + 8*ti

    if (nChunk > 0) {
        {   // prime: chunk 0 -> buffer 0
            int e = kbeg + srow; if (e >= kend) e = kend - 1;
            const unsigned la = lds0 + (unsigned)((srow * SSTRIDE + sseg * 32) * 4);
            const unsigned long long ga =
                (unsigned long long)(uintptr_t)(xsrc + (size_t)scol[e] * NG + sseg * 32);
#pragma unroll
            for (int q = 0; q < 8; ++q) async_ld_b128(la + q * 16, ga + q * 16);
        }
        for (int ci = 0; ci < nChunk; ++ci) {
            const int buf = ci & 1;
            const int sbase = buf * CHUNK * SSTRIDE;
            if (ci + 1 < nChunk) {      // issue next chunk into other buffer
                int e = kbeg + (ci + 1) * CHUNK + srow; if (e >= kend) e = kend - 1;
                const unsigned la = lds0 +
                    (unsigned)((((buf ^ 1) * CHUNK + srow) * SSTRIDE + sseg * 32) * 4);
                const unsigned long long ga =
                    (unsigned long long)(uintptr_t)(xsrc + (size_t)scol[e] * NG + sseg * 32);
#pragma unroll
                for (int q = 0; q < 8; ++q) async_ld_b128(la + q * 16, ga + q * 16);
            }
            if (t < CHUNK) {            // local row + weight of chunk ci's edges
                int e = kbeg + ci * CHUNK + t;
                float2 rec;
                if (e < kend) {
                    int lr = 0;
#pragma unroll
                    for (int j = 1; j < RPB; ++j) if (e >= rp[j]) lr = j;
                    rec.x = swt[e]; rec.y = __int_as_float(lr);
                } else { rec.x = 0.f; rec.y = __int_as_float(0); }
                lw2[t] = rec;
            }
            if (ci + 1 < nChunk) { WAIT_ASYNC(8); } else { WAIT_ASYNC(0); }
            __syncthreads();

#pragma unroll
            for (int s = 0; s < CHUNK / 4; ++s) {
                // A (16x4, M=dest row, K=edge): lanes 0-15 carry K0/K1, lanes 16-31 K2/K3
                // branch-free: unconditional b64 record loads + arithmetic select
                const float2 e0 = lw2[4 * s + 2 * half];
                const float2 e1 = lw2[4 * s + 2 * half + 1];
                const float a0 = e0.x * (float)(__float_as_int(e0.y) == lm);
                const float a1 = e1.x * (float)(__float_as_int(e1.y) == lm);
                const v2f av = {a0, a1};
                // B (4x16, K=edge, N=gene): VGPR0 carries K0(lo)/K1(hi), VGPR1 K2/K3
                const int eB0 = 4 * s + half, eB1 = eB0 + 2;
#pragma unroll
                for (int ti = 0; ti < 4; ++ti) {
                    const int gb = (wv + 8 * ti) * 16;
                    const float b0 = stageMem[sbase + eB0 * SSTRIDE + gb + lm];
                    const float b1 = stageMem[sbase + eB1 * SSTRIDE + gb + lm];
                    const v2f bv = {b0, b1};
                    dsum[ti] = __builtin_amdgcn_wmma_f32_16x16x4_f32(
                        false, av, false, bv, (short)0, dsum[ti], false, false);
                }
            }
            __syncthreads();            // release buffer `buf` for reuse
        }
    }

    // ---- spill D tiles to LDS xp (reuse stage memory) -------------------
    __syncthreads();
    float* xp = stageMem;               // 16 x 512 floats
#pragma unroll
    for (int ti = 0; ti < 4; ++ti) {
        const int gb = (wv + 8 * ti) * 16;
        const float* dp = (const float*)&dsum[ti];
#pragma unroll
        for (int v = 0; v < 8; ++v)
            xp[(v + 8 * half) * NG + gb + lm] = dp[v];   // D: lane half -> M=v / v+8
    }
    __syncthreads();

    // ---- per-row mask merge + capped random selection -------------------
    const int g0 = 2 * t, g1 = 2 * t + 1;
    for (int rr = 0; rr < RPB; ++rr) {
        const int r = r_base + rr;
        const float acc0 = xp[rr * NG + g0];
        const float acc1 = xp[rr * NG + g1];

        const float* xo = xorig + (size_t)r * NG;
        unsigned char* impRow = imp + (size_t)r * NG;

        const float2 o = *(const float2*)(xo + g0);
        const bool ex0 = (o.x != 0.f), ex1 = (o.y != 0.f);
        float v0 = ex0 ? o.x : acc0;
        float v1 = ex1 ? o.y : acc1;

        const uchar2 ip = *(const uchar2*)(impRow + g0);
        const bool im0 = (ip.x != 0) || (v0 != 0.f && !ex0);
        const bool im1 = (ip.y != 0) || (v1 != 0.f && !ex1);

        const unsigned rg0 = (unsigned)r * NG + (unsigned)g0;
        const unsigned h0  = im0 ? rnd_hash((unsigned)it, rg0)      : 0xFFFFFFFFu;
        const unsigned h1  = im1 ? rnd_hash((unsigned)it, rg0 + 1u) : 0xFFFFFFFFu;
        sc[g0] = h0; sc[g1] = h1;
        hist[t] = 0;
        __syncthreads();

        if (im0) atomicAdd(&hist[h0 >> 24], 1u);
        if (im1) atomicAdd(&hist[h1 >> 24], 1u);
        __syncthreads();

        const unsigned mycnt = hist[t];
        for (int off = 1; off < 256; off <<= 1) {
            unsigned a = (t >= off) ? hist[t - off] : 0u;
            __syncthreads();
            hist[t] += a;
            __syncthreads();
        }
        cumex[t] = (int)(hist[t] - mycnt);
        if (t == 255) sTotal = (int)hist[255];
        const int cap = caps[r];
        if ((int)hist[t] >= cap && (t == 0 || (int)hist[t - 1] < cap)) sT = t;
        __syncthreads();

        const int total = sTotal;
        bool k0 = false, k1 = false;
        if (im0) {
            if (total <= cap) k0 = true;
            else {
                const int T = sT, b0i = (int)(h0 >> 24);
                if (b0i < T) k0 = true;
                else if (b0i == T) {
                    const int quota = cap - cumex[T];
                    int c2 = 0;
                    for (int j = 0; j < NG; ++j) {
                        unsigned s = sc[j];
                        if ((int)(s >> 24) == T && (s < h0 || (s == h0 && j < g0))) c2++;
                    }
                    k0 = (c2 < quota);
                }
            }
        }
        if (im1) {
            if (total <= cap) k1 = true;
            else {
                const int T = sT, b1i = (int)(h1 >> 24);
                if (b1i < T) k1 = true;
                else if (b1i == T) {
                    const int quota = cap - cumex[T];
                    int c2 = 0;
                    for (int j = 0; j < NG; ++j) {
                        unsigned s = sc[j];
                        if ((int)(s >> 24) == T && (s < h1 || (s == h1 && j < g1))) c2++;
                    }
                    k1 = (c2 < quota);
                }
            }
        }

        if (im0 && !k0) v0 = 0.f;
        if (im1 && !k1) v1 = 0.f;

        uchar2 ipn; ipn.x = (unsigned char)(im0 && k0); ipn.y = (unsigned char)(im1 && k1);
        *(uchar2*)(impRow + g0) = ipn;

        float2 vo; vo.x = v0; vo.y = v1;
        *(float2*)(xdst + (size_t)r * NG + g0) = vo;
        __syncthreads();                // hist/sc reuse for next row
    }
}

// ---- launcher -----------------------------------------------------------
extern "C" void kernel_launch(void* const* d_in, const int* in_sizes, int n_in,
                              void* d_out, int out_size, void* d_ws, size_t ws_size,
                              hipStream_t stream) {
    (void)n_in; (void)out_size; (void)ws_size;
    const float* xin  = (const float*)d_in[0];
    const float* ew   = (const float*)d_in[1];
    const int*   ei   = (const int*)d_in[2];
    const int*   caps = (const int*)d_in[3];
    const int E = in_sizes[1];
    const int N = in_sizes[3];

    char* ws = (char*)d_ws;
    int*           rowPtr = (int*)(ws + OFF_ROWPTR);
    int*           cursor = (int*)(ws + OFF_CURSOR);
    int*           scol   = (int*)(ws + OFF_SCOL);
    float*         swt    = (float*)(ws + OFF_SWT);
    unsigned char* imp    = (unsigned char*)(ws + OFF_IMP);
    float*         xbuf   = (float*)(ws + OFF_XBUF);

    const int* rows = ei;
    const int* cols = ei + E;

    k_zero<<<1024, 256, 0, stream>>>(cursor, (size_t)N);
    k_zero<<<2048, 256, 0, stream>>>((int*)imp, ((size_t)N * NG) / 4);
    k_hist<<<(E + 255) / 256, 256, 0, stream>>>(rows, cursor, E);
    k_scan<<<1, 1024, 0, stream>>>(cursor, rowPtr, N);
    k_scatter<<<(E + 255) / 256, 256, 0, stream>>>(rows, cols, ew, cursor, scol, swt, E);
    k_sortrows<<<(N + 255) / 256, 256, 0, stream>>>(rowPtr, scol, swt, N);

    const float* src = xin;
    float*       dst = (float*)d_out;
    for (int it = 0; it < NITER; ++it) {
        k_prop<<<N / RPB, 256, 0, stream>>>(src, dst, xin, imp, rowPtr, scol, swt, caps, it);
        const float* nsrc = dst;
        dst = (dst == (float*)d_out) ? xbuf : (float*)d_out;
        src = nsrc;
    }
}